// CFSSDDefaultLoss_61821759259089
// MI455X (gfx1250) — compile-verified
//
#include <hip/hip_runtime.h>
#include <math.h>

// Problem constants (from reference)
#define BB 32
#define AA 8732
#define CC 201
#define GG 50

typedef __attribute__((ext_vector_type(2))) float v2f;
typedef __attribute__((ext_vector_type(8))) float v8f;

// ---------------------------------------------------------------------------
// Kernel 1: smooth-L1 bbox loss + foreground counts (per image, atomics)
// ---------------------------------------------------------------------------
__global__ void __launch_bounds__(256)
k_bbox(const float* __restrict__ boxes, const float* __restrict__ bbox_reg,
       const float* __restrict__ anchors, const int* __restrict__ matched,
       float* __restrict__ ws_bbox, int* __restrict__ ws_fgcnt) {
    const int b = blockIdx.y;
    const int a = blockIdx.x * blockDim.x + threadIdx.x;
    float s = 0.f;
    int cnt = 0;
    if (a < AA) {
        const int m = matched[b * AA + a];
        if (m >= 0) {
            cnt = 1;
            const float4 anc = ((const float4*)anchors)[b * AA + a];
            const float4 gt  = ((const float4*)boxes)[b * GG + m];
            const float4 pr  = ((const float4*)bbox_reg)[b * AA + a];
            const float aw = anc.z - anc.x, ah = anc.w - anc.y;
            const float acx = anc.x + 0.5f * aw, acy = anc.y + 0.5f * ah;
            const float gw = gt.z - gt.x, gh = gt.w - gt.y;
            const float gcx = gt.x + 0.5f * gw, gcy = gt.y + 0.5f * gh;
            const float t0 = 10.f * (gcx - acx) / aw;
            const float t1 = 10.f * (gcy - acy) / ah;
            const float t2 = 5.f * logf(gw / aw);
            const float t3 = 5.f * logf(gh / ah);
            float d, ad;
            d = pr.x - t0; ad = fabsf(d); s += (ad < 1.f) ? 0.5f * d * d : ad - 0.5f;
            d = pr.y - t1; ad = fabsf(d); s += (ad < 1.f) ? 0.5f * d * d : ad - 0.5f;
            d = pr.z - t2; ad = fabsf(d); s += (ad < 1.f) ? 0.5f * d * d : ad - 0.5f;
            d = pr.w - t3; ad = fabsf(d); s += (ad < 1.f) ? 0.5f * d * d : ad - 0.5f;
        }
    }
    __shared__ float sf[256];
    __shared__ int   si[256];
    sf[threadIdx.x] = s;
    si[threadIdx.x] = cnt;
    __syncthreads();
    for (int off = 128; off > 0; off >>= 1) {
        if (threadIdx.x < off) {
            sf[threadIdx.x] += sf[threadIdx.x + off];
            si[threadIdx.x] += si[threadIdx.x + off];
        }
        __syncthreads();
    }
    if (threadIdx.x == 0) {
        atomicAdd(&ws_bbox[b], sf[0]);
        atomicAdd(&ws_fgcnt[b], si[0]);
    }
}

// ---------------------------------------------------------------------------
// Kernel 2 (the hot one): per-anchor cross-entropy.
// 128 threads = 4 waves / block; each wave owns 16 anchors.
//  - Staging: GLOBAL_LOAD_ASYNC_TO_LDS_B32 (ASYNCcnt) — direct global->LDS,
//    no VGPR round trip. Tail rows are clamped (whole rows, results discarded).
//  - Sum-exp row reduction on the matrix pipe: WMMA f32 16x16x4 with B = ones
//    gives exact fp32 row sums and co-executes with v_exp_f32.
//  - Steady-state loop is branchless: 50 full K-chunks + 1 tail chunk.
// ---------------------------------------------------------------------------
__global__ void __launch_bounds__(128)
k_ce(const float* __restrict__ cls_logits, const int* __restrict__ labels,
     const int* __restrict__ matched,
     float* __restrict__ ws_cls, float* __restrict__ ws_fgcls) {
    const int b = blockIdx.y;
    const int base = blockIdx.x * 64;

    __shared__ float tile[64 * CC];      // 51,456 B
    __shared__ float rowsum[4][16];

    // --- Async global->LDS staging of 64 contiguous anchor rows -------------
    {
        const float* gsrc = cls_logits + (long)b * AA * CC;
        const unsigned ldsbase = (unsigned)(size_t)(&tile[0]);
        for (int i = threadIdx.x; i < 64 * CC; i += 128) {
            int ai = base * CC + i;
            if (ai >= AA * CC) ai = AA * CC - 1;      // clamp: rows discarded anyway
            const unsigned long long gaddr = (unsigned long long)(size_t)(gsrc + ai);
            const unsigned laddr = ldsbase + (unsigned)i * 4u;
            asm volatile("global_load_async_to_lds_b32 %0, %1, off"
                         :: "v"(laddr), "v"(gaddr)
                         : "memory");
        }
        asm volatile("s_wait_asynccnt 0x0" ::: "memory");
    }
    __syncthreads();

    const int lane = threadIdx.x & 31;
    const int w    = threadIdx.x >> 5;
    const int M    = lane & 15;          // anchor-in-wave (A-matrix row)
    const int half = lane >> 4;          // K sub-lane group
    const float* row = &tile[(w * 16 + M) * CC];

    // Pass 1: row max — 50 unguarded chunks (K=0..199), branchless tail (K=200)
    float vmax = -__builtin_inff();
    for (int c = 0; c < 200; c += 4) {
        const int k0 = c + 2 * half;
        vmax = fmaxf(vmax, row[k0]);
        vmax = fmaxf(vmax, row[k0 + 1]);
    }
    {
        const float t = row[200];
        vmax = fmaxf(vmax, (half == 0) ? t : -__builtin_inff());
    }
    vmax = fmaxf(vmax, __shfl_xor(vmax, 16));

    // Pass 2: sum of exp via WMMA accumulation, B = all-ones 4x16.
    v8f acc = {0.f, 0.f, 0.f, 0.f, 0.f, 0.f, 0.f, 0.f};
    v2f bones; bones.x = 1.f; bones.y = 1.f;
    for (int c = 0; c < 200; c += 4) {
        const int k0 = c + 2 * half;
        v2f amat;
        amat.x = __expf(row[k0]     - vmax);
        amat.y = __expf(row[k0 + 1] - vmax);
        acc = __builtin_amdgcn_wmma_f32_16x16x4_f32(
            false, amat, false, bones, (short)0, acc, false, false);
    }
    {   // tail chunk: only K=200 (lanes 0-15) contributes
        v2f amat;
        const float e = __expf(row[200] - vmax);
        amat.x = (half == 0) ? e : 0.f;
        amat.y = 0.f;
        acc = __builtin_amdgcn_wmma_f32_16x16x4_f32(
            false, amat, false, bones, (short)0, acc, false, false);
    }

    // D layout: VGPR r holds M=r (lanes 0-15) / M=8+r (lanes 16-31); any column.
    if (lane == 0 || lane == 16) {
        const int mbase = (lane == 0) ? 0 : 8;
        #pragma unroll
        for (int r = 0; r < 8; ++r) rowsum[w][mbase + r] = acc[r];
    }
    __syncthreads();

    if (lane < 16) {                     // lane == anchor-in-wave; has full vmax
        const int a = base + w * 16 + M;
        if (a < AA) {
            const int mi  = matched[b * AA + a];
            const int tgt = (mi >= 0) ? labels[b * GG + mi] : 0;
            const float loss = vmax + logf(rowsum[w][M]) - row[tgt];
            ws_cls[b * AA + a] = loss;
            if (mi >= 0) atomicAdd(&ws_fgcls[b], loss);
        }
    }
}

// ---------------------------------------------------------------------------
// Kernel 3: hard-negative mining = sum of top-k negative losses per image.
// Exact k-th-largest via 31-step bit-space binary search (values >= 0 so
// float bits are monotone); ties are equal-valued so the sum matches the
// reference's stable double-argsort selection.
// ---------------------------------------------------------------------------
__global__ void __launch_bounds__(256)
k_topk(const int* __restrict__ matched, const int* __restrict__ ws_fgcnt,
       const float* __restrict__ ws_cls, float* __restrict__ ws_bg) {
    const int b = blockIdx.x;
    __shared__ float vals[AA];           // 34,928 B
    __shared__ unsigned scnt;
    __shared__ float ssum;

    for (int i = threadIdx.x; i < AA; i += 256) {
        const float v = ws_cls[b * AA + i];
        vals[i] = (matched[b * AA + i] >= 0) ? -1.f : v;   // fg excluded (< 0)
    }
    const int k = 3 * ws_fgcnt[b];
    __syncthreads();
    if (k <= 0) {
        if (threadIdx.x == 0) ws_bg[b] = 0.f;
        return;
    }

    unsigned P = 0;   // largest float bits with count(v > P) >= k
    for (int bit = 30; bit >= 0; --bit) {
        const unsigned cand = P | (1u << bit);
        const float cf = __uint_as_float(cand);
        if (threadIdx.x == 0) scnt = 0;
        __syncthreads();
        unsigned local = 0;
        for (int i = threadIdx.x; i < AA; i += 256) local += (vals[i] > cf);
        if (local) atomicAdd(&scnt, local);
        __syncthreads();
        const unsigned c = scnt;
        __syncthreads();
        if (c >= (unsigned)k) P = cand;
    }
    const float T = __uint_as_float(P + 1u);   // == k-th largest value

    if (threadIdx.x == 0) { scnt = 0; ssum = 0.f; }
    __syncthreads();
    unsigned lc = 0; float ls = 0.f;
    for (int i = threadIdx.x; i < AA; i += 256) {
        const float v = vals[i];
        if (v >= T) { lc++; ls += v; }
    }
    if (lc) { atomicAdd(&scnt, lc); atomicAdd(&ssum, ls); }
    __syncthreads();
    if (threadIdx.x == 0)
        ws_bg[b] = ssum - (float)((int)scnt - k) * T;
}

// ---------------------------------------------------------------------------
// Kernel 4: final scalar combine + validation head (trivial work).
// ---------------------------------------------------------------------------
__global__ void k_final(const float* __restrict__ rej, const int* __restrict__ img_label,
                        const float* __restrict__ ws_bbox, const float* __restrict__ ws_fgcls,
                        const float* __restrict__ ws_bg, const int* __restrict__ ws_fgcnt,
                        float* __restrict__ out) {
    if (threadIdx.x != 0 || blockIdx.x != 0) return;
    int nfg = 0; float breg = 0.f, fcls = 0.f, bcls = 0.f;
    for (int b = 0; b < BB; ++b) {
        nfg  += ws_fgcnt[b];
        breg += ws_bbox[b];
        fcls += ws_fgcls[b];
        bcls += ws_bg[b];
    }
    const float N = (float)((nfg > 1) ? nfg : 1);
    const float reg_loss = breg / N;
    const float cls_loss = (fcls + bcls) / N;     // N_neg == 0 (MIN_NEG_SAMPLES=0)
    float vsum = 0.f;
    for (int b = 0; b < BB; ++b) {
        const float l0 = rej[2 * b], l1 = rej[2 * b + 1];
        const float m = fmaxf(l0, l1);
        const float lse = m + logf(expf(l0 - m) + expf(l1 - m));
        vsum += lse - (img_label[b] ? l1 : l0);
    }
    const float val_loss = vsum / (float)BB;
    out[0] = 0.5f * (reg_loss + cls_loss) + 0.5f * val_loss;
    out[1] = reg_loss;
    out[2] = cls_loss;
    out[3] = val_loss;
}

// ---------------------------------------------------------------------------
extern "C" void kernel_launch(void* const* d_in, const int* in_sizes, int n_in,
                              void* d_out, int out_size, void* d_ws, size_t ws_size,
                              hipStream_t stream) {
    const float* boxes       = (const float*)d_in[0];
    const int*   labels      = (const int*)  d_in[1];
    const int*   image_label = (const int*)  d_in[2];
    const float* bbox_reg    = (const float*)d_in[3];
    const float* cls_logits  = (const float*)d_in[4];
    const float* rej         = (const float*)d_in[5];
    const float* anchors     = (const float*)d_in[6];
    const int*   matched     = (const int*)  d_in[7];
    float* out = (float*)d_out;

    // Workspace layout (all 32-bit words)
    float* ws_cls   = (float*)d_ws;          // B*A per-anchor CE loss
    float* ws_bbox  = ws_cls + BB * AA;      // B
    float* ws_fgcls = ws_bbox + BB;          // B
    float* ws_bg    = ws_fgcls + BB;         // B
    int*   ws_fgcnt = (int*)(ws_bg + BB);    // B

    // Zero the 4*B accumulator words (graph-capture-safe async memset)
    hipMemsetAsync(ws_bbox, 0, 4 * BB * sizeof(float), stream);

    dim3 g2((AA + 255) / 256, BB);
    k_bbox<<<g2, 256, 0, stream>>>(boxes, bbox_reg, anchors, matched, ws_bbox, ws_fgcnt);

    dim3 g3((AA + 63) / 64, BB);
    k_ce<<<g3, 128, 0, stream>>>(cls_logits, labels, matched, ws_cls, ws_fgcls);

    k_topk<<<BB, 256, 0, stream>>>(matched, ws_fgcnt, ws_cls, ws_bg);

    k_final<<<1, 64, 0, stream>>>(rej, image_label, ws_bbox, ws_fgcls, ws_bg, ws_fgcnt, out);
}